// YoloPredict_61460982005913
// MI455X (gfx1250) — compile-verified
//
#include <hip/hip_runtime.h>
#include <cstdint>

// ---------------------------------------------------------------------------
// YOLO region post-processing for MI455X (gfx1250, wave32).
//   Kernel 1: elementwise decode (sigmoid/softmax/exp) -> class probs, boxes,
//             thresholded max-score (stored temporarily in prob[...,20]).
//   Kernel 2: fused per-batch NMS: bitonic sort (LDS) -> tiled suppression
//             mask (16x16 tiles; union outer-sum on the WMMA matrix pipe via
//             V_WMMA_F32_16X16X4_F32, intersection min/max on VALU) -> serial
//             greedy bit-resolve -> scatter keep*score back to original order.
// ---------------------------------------------------------------------------

#define NCLS   20
#define NANC   5
#define FHH    19
#define FWW    19
#define BATCH  32
#define NBOX   (NANC * FHH * FWW)   // 1805
#define NTILE  113                  // ceil(1805/16)
#define NPAD   (NTILE * 16)         // 1808
#define NSORT  2048
#define NMS_T  0.45f
#define PRE_T  0.005f

typedef float v2f __attribute__((ext_vector_type(2)));
typedef float v8f __attribute__((ext_vector_type(8)));

__device__ __forceinline__ float sigmoidf(float z) {
    return 1.0f / (1.0f + expf(-z));
}

// ---------------------------------------------------------------------------
// Kernel 1: decode. One thread per (b, a, h, w) cell.
// ---------------------------------------------------------------------------
__global__ void yolo_decode(const float* __restrict__ x,
                            const float* __restrict__ im_info,
                            const float* __restrict__ biases,
                            float* __restrict__ out) {
    int t = blockIdx.x * blockDim.x + threadIdx.x;
    if (t >= BATCH * NBOX) return;
    int b = t / NBOX, n = t % NBOX;
    int a = n / (FHH * FWW), rem = n % (FHH * FWW);
    int h = rem / FWW, w = rem % FWW;
    const int HW = FHH * FWW;
    const float* xb = x + (size_t)b * 125 * HW + h * FWW + w;  // channel c at xb[c*HW]

    // softmax over the 20 class logits (channels 25 + a*20 + c)
    float conf[NCLS];
    float m = -3.4e38f;
    #pragma unroll
    for (int c = 0; c < NCLS; ++c) {
        conf[c] = xb[(25 + a * NCLS + c) * HW];
        m = fmaxf(m, conf[c]);
    }
    float s = 0.f;
    #pragma unroll
    for (int c = 0; c < NCLS; ++c) { conf[c] = expf(conf[c] - m); s += conf[c]; }
    float inv = 1.0f / s;
    float obj = sigmoidf(xb[(20 + a) * HW]);

    float* prob = out + ((size_t)b * NBOX + n) * (NCLS + 1);
    float mp = 0.f;
    #pragma unroll
    for (int c = 0; c < NCLS; ++c) {
        float p = conf[c] * inv * obj;   // class_prob * objectness
        prob[c] = p;
        mp = fmaxf(mp, p);
    }
    float score = (mp > PRE_T) ? mp : 0.f;   // pre_threshold
    prob[NCLS] = score;                      // temp; NMS kernel overwrites

    // darknet box decode -> image coords (cx, cy, w, h)
    float im_h = im_info[2 * b + 0];
    float im_w = im_info[2 * b + 1];
    float xy0 = sigmoidf(xb[(2 * a + 0) * HW]);
    float xy1 = sigmoidf(xb[(2 * a + 1) * HW]);
    float wh0 = xb[(10 + 2 * a + 0) * HW];
    float wh1 = xb[(10 + 2 * a + 1) * HW];
    float bx = (xy0 + (float)w) / (float)FWW * im_w;
    float by = (xy1 + (float)h) / (float)FHH * im_h;
    float bw = expf(wh0) * biases[2 * a + 0] / (float)FWW * im_w;
    float bh = expf(wh1) * biases[2 * a + 1] / (float)FHH * im_h;
    float* boxes = out + (size_t)BATCH * NBOX * (NCLS + 1) + ((size_t)b * NBOX + n) * 4;
    boxes[0] = bx; boxes[1] = by; boxes[2] = bw; boxes[3] = bh;
}

// ---------------------------------------------------------------------------
// Kernel 2: fused per-batch sort + mask-matrix NMS. 1 block = 1 batch.
// ---------------------------------------------------------------------------
struct SMem {
    union {
        unsigned long long keys[NSORT];                     // 16384 B (sort phase)
        struct {
            unsigned rowbits[16][NTILE];                    //  7232 B (mask tile bits)
            unsigned sup[NPAD];                             //  7232 B (suppressed flags)
        } r;
    } u;
    float4         sbox[NPAD];     // sorted boxes as (x1,y1,x2,y2)   28928 B
    float          sarea[NPAD];    // sorted areas                      7232 B
    float          sscore[NPAD];   // sorted scores                     7232 B
    unsigned short sidx[NPAD];     // original index per sorted slot    3616 B
};                                 // total 63392 B < 64 KB static LDS

__global__ __launch_bounds__(1024) void yolo_nms(float* __restrict__ out) {
    const int b    = blockIdx.x;
    const int tid  = threadIdx.x;
    const int lane = tid & 31;
    const int wv   = tid >> 5;   // wave index within block (32 waves)

    __shared__ SMem S;

    float* prob = out;
    const float* boxes = out + (size_t)BATCH * NBOX * (NCLS + 1);

    // ---- 1. build 64-bit sort keys: (score_bits << 32) | (0x7FFFFFFF - n) ----
    for (int k = tid; k < NSORT; k += 1024) {
        unsigned long long key = 0ull;
        if (k < NBOX) {
            float sc = prob[((size_t)b * NBOX + k) * (NCLS + 1) + NCLS];  // >= 0
            unsigned sb = __float_as_uint(sc);
            key = ((unsigned long long)sb << 32) |
                  (unsigned long long)(0x7FFFFFFFu - (unsigned)k);
        }
        S.u.keys[k] = key;
    }
    __syncthreads();

    // ---- 2. bitonic sort, descending (stable via index in low bits) ----
    for (int kk = 2; kk <= NSORT; kk <<= 1) {
        for (int j = kk >> 1; j > 0; j >>= 1) {
            for (int t = tid; t < NSORT; t += 1024) {
                int p = t ^ j;
                if (p > t) {
                    unsigned long long ka = S.u.keys[t], kb = S.u.keys[p];
                    bool desc = ((t & kk) == 0);
                    if (desc ? (ka < kb) : (ka > kb)) {
                        S.u.keys[t] = kb; S.u.keys[p] = ka;
                    }
                }
            }
            __syncthreads();
        }
    }

    // ---- 3. gather sorted boxes/areas/scores/indices (pad to 1808) ----
    for (int k = tid; k < NPAD; k += 1024) {
        unsigned long long key = S.u.keys[k];
        unsigned n = 0x7FFFFFFFu - (unsigned)(key & 0xFFFFFFFFull);
        float4 bb; float area, sc; unsigned short si;
        if (n < NBOX) {
            const float* bp = boxes + ((size_t)b * NBOX + n) * 4;
            float cx = bp[0], cy = bp[1], ww = bp[2], hh = bp[3];
            bb.x = cx - ww * 0.5f; bb.y = cy - hh * 0.5f;
            bb.z = cx + ww * 0.5f; bb.w = cy + hh * 0.5f;
            area = ww * hh;
            sc = __uint_as_float((unsigned)(key >> 32));
            si = (unsigned short)n;
        } else {
            bb.x = bb.y = bb.z = bb.w = -1.0e30f;  // degenerate: inter==0, area==0
            area = 0.f; sc = 0.f; si = 0xFFFFu;
        }
        S.sbox[k] = bb; S.sarea[k] = area; S.sscore[k] = sc; S.sidx[k] = si;
    }
    __syncthreads();

    // ---- 4. clear suppression flags (reuses the dead sort-key region) ----
    for (int k = tid; k < NPAD; k += 1024) S.u.r.sup[k] = 0u;
    __syncthreads();

    // ---- 5. tile loop: parallel 16xN mask tiles, then serial row resolve ----
    for (int ti = 0; ti < NTILE; ++ti) {
        // WMMA A operand (16x4 f32): row m = [area_i[m], 1, 0, 0]
        // lanes 0-15 hold K=0 (V0) and K=1 (V1); lanes 16-31 hold K=2,3 (zeros)
        v2f av;
        av.x = (lane < 16) ? S.sarea[ti * 16 + lane] : 0.f;
        av.y = (lane < 16) ? 1.0f : 0.f;
        const int base = (lane < 16) ? 0 : 8;   // C/D rows: v -> M=v (lo) / v+8 (hi)
        float4 rbox[8];
        #pragma unroll
        for (int v = 0; v < 8; ++v) rbox[v] = S.sbox[ti * 16 + base + v];

        for (int tj = ti + wv; tj < NTILE; tj += 32) {     // wave-uniform loop
            int jn = tj * 16 + (lane & 15);
            float4 cb = S.sbox[jn];
            // WMMA B operand (4x16 f32): col n = [1, area_j[n], 0, 0]
            v2f bv;
            bv.x = (lane < 16) ? 1.0f : 0.f;
            bv.y = (lane < 16) ? S.sarea[jn] : 0.f;
            v8f acc = {};
#if __has_builtin(__builtin_amdgcn_wmma_f32_16x16x4_f32)
            // acc[m][n] = area_i[m] + area_j[n]  (rank-2 outer-sum on matrix pipe)
            acc = __builtin_amdgcn_wmma_f32_16x16x4_f32(
                false, av, false, bv, (short)0, acc, false, false);
#else
            #pragma unroll
            for (int v = 0; v < 8; ++v)
                acc[v] = S.sarea[ti * 16 + base + v] + S.sarea[jn];
#endif
            unsigned ball[8];
            #pragma unroll
            for (int v = 0; v < 8; ++v) {
                float ix = fminf(rbox[v].z, cb.z) - fmaxf(rbox[v].x, cb.x);
                float iy = fminf(rbox[v].w, cb.w) - fmaxf(rbox[v].y, cb.y);
                ix = fmaxf(ix, 0.f); iy = fmaxf(iy, 0.f);
                float inter = ix * iy;
                float uni = acc[v] - inter;           // area_i + area_j - inter
                int ig = ti * 16 + base + v;
                int jg = tj * 16 + (lane & 15);
                bool pred = (inter > NMS_T * uni) && (jg > ig);
                ball[v] = __builtin_amdgcn_ballot_w32(pred);
            }
            // pack: ballot bits [15:0] = row v cols 0-15, [31:16] = row v+8
            if (lane < 16) {
                unsigned sel = 0;
                #pragma unroll
                for (int v = 0; v < 8; ++v) sel = ((lane & 7) == v) ? ball[v] : sel;
                unsigned bits = (lane < 8) ? (sel & 0xFFFFu) : ((sel >> 16) & 0xFFFFu);
                S.u.r.rowbits[lane][tj] = bits;
            }
        }
        __syncthreads();

        // serial greedy resolve of the 16 rows of this i-tile
        for (int r = 0; r < 16; ++r) {
            int i = ti * 16 + r;
            bool act = (S.u.r.sup[i] == 0u) && (S.sscore[i] > 0.f);  // block-uniform
            if (act) {
                for (int jg = ti * 16 + tid; jg < NPAD; jg += 1024) {
                    unsigned bits = S.u.r.rowbits[r][jg >> 4];
                    if ((bits >> (jg & 15)) & 1u) S.u.r.sup[jg] = 1u;  // bits imply jg > i
                }
            }
            __syncthreads();
        }
    }

    // ---- 6. keep mask & scatter nms_score back to original order ----
    for (int k = tid; k < NPAD; k += 1024) {
        unsigned n = S.sidx[k];
        if (n < NBOX) {
            float sc = S.sscore[k];
            float v = (S.u.r.sup[k] == 0u && sc > 0.f) ? sc : 0.f;
            prob[((size_t)b * NBOX + n) * (NCLS + 1) + NCLS] = v;
        }
    }
}

// ---------------------------------------------------------------------------
extern "C" void kernel_launch(void* const* d_in, const int* in_sizes, int n_in,
                              void* d_out, int out_size, void* d_ws, size_t ws_size,
                              hipStream_t stream) {
    (void)in_sizes; (void)n_in; (void)out_size; (void)d_ws; (void)ws_size;
    const float* x       = (const float*)d_in[0];
    const float* im_info = (const float*)d_in[1];
    const float* biases  = (const float*)d_in[2];
    float* out = (float*)d_out;

    int total = BATCH * NBOX;
    yolo_decode<<<(total + 255) / 256, 256, 0, stream>>>(x, im_info, biases, out);
    yolo_nms<<<BATCH, 1024, 0, stream>>>(out);
}